// WaveConv3d_5600637354125
// MI455X (gfx1250) — compile-verified
//
#include <hip/hip_runtime.h>
#include <hip/hip_bf16.h>

// ---------------------------------------------------------------------------
// WaveConv3d for MI455X (gfx1250), fp32 throughout.
//   K1: fused 2-level Haar analysis  -> S[2][8][4][4096][32]   (32 MiB ws)
//   K2: per-voxel channel mix via V_WMMA_F32_16X16X4_F32
//       (M=4 batch, N=32 out-ch, K=64 concat in-ch)            -> Mixed 16 MiB
//   K3: fused 1-level Haar synthesis + zero-detail upsample    -> out 128 MiB
// Bandwidth-bound (~640 MiB min traffic @ 23.3 TB/s ~ 27 us); all HBM-facing
// accesses are coalesced; weights staged via LDS with b128 moves.
// ---------------------------------------------------------------------------

typedef __attribute__((ext_vector_type(2))) float v2f;
typedef __attribute__((ext_vector_type(8))) float v8f;

// ------------------------------ Kernel 1: forward 2-level Haar --------------
// thread = (src, b, c, Z, Y, X); reads a 4x4x4 block of x, writes 8 subbands.
__global__ __launch_bounds__(256) void wave_dwt_kernel(
    const float* __restrict__ x1, const float* __restrict__ x2,
    float* __restrict__ S)  // [src 2][k 8][b 4][p 4096][c 32]
{
    const int t = blockIdx.x * 256 + threadIdx.x;      // 0 .. 2^20-1
    const int X = t & 15, Y = (t >> 4) & 15, Z = (t >> 8) & 15;
    const int c = (t >> 12) & 31, b = (t >> 17) & 3, src = t >> 19;
    const int p = t & 4095;                            // Z*256 + Y*16 + X

    const float* x = src ? x2 : x1;
    const float* base = x + ((size_t)(b * 32 + c) << 18);   // * 64*64*64

    // sums of the eight fine 2x2x2 blocks inside the 4x4x4 block
    float A1[2][2][2] = {};
#pragma unroll
    for (int iz = 0; iz < 4; ++iz) {
#pragma unroll
        for (int iy = 0; iy < 4; ++iy) {
            const float4 v = *(const float4*)(
                base + (((size_t)(4 * Z + iz) * 64 + (4 * Y + iy)) * 64 + 4 * X));
            A1[iz >> 1][iy >> 1][0] += v.x + v.y;
            A1[iz >> 1][iy >> 1][1] += v.z + v.w;
        }
    }
    // sign-Hadamard over (dx, dy, dz):  s_k = (1/8) * sum_d (-1)^(d.k) A1[d]
    float t0[2][2][2], t1[2][2][2], t2[2][2][2];
#pragma unroll
    for (int dz = 0; dz < 2; ++dz)
#pragma unroll
        for (int dy = 0; dy < 2; ++dy) {
            t0[dz][dy][0] = A1[dz][dy][0] + A1[dz][dy][1];
            t0[dz][dy][1] = A1[dz][dy][0] - A1[dz][dy][1];
        }
#pragma unroll
    for (int dz = 0; dz < 2; ++dz)
#pragma unroll
        for (int kx = 0; kx < 2; ++kx) {
            t1[dz][0][kx] = t0[dz][0][kx] + t0[dz][1][kx];
            t1[dz][1][kx] = t0[dz][0][kx] - t0[dz][1][kx];
        }
#pragma unroll
    for (int ky = 0; ky < 2; ++ky)
#pragma unroll
        for (int kx = 0; kx < 2; ++kx) {
            t2[0][ky][kx] = t1[0][ky][kx] + t1[1][ky][kx];
            t2[1][ky][kx] = t1[0][ky][kx] - t1[1][ky][kx];
        }
#pragma unroll
    for (int kz = 0; kz < 2; ++kz)
#pragma unroll
        for (int ky = 0; ky < 2; ++ky)
#pragma unroll
            for (int kx = 0; kx < 2; ++kx) {
                const int k = kz * 4 + ky * 2 + kx;
                S[((size_t)((src * 8 + k) * 4 + b) * 4096 + p) * 32 + c] =
                    0.125f * t2[kz][ky][kx];
            }
}

// ------------------------------ Kernel 2: WMMA channel mix ------------------
// workgroup = (subband k, 16 consecutive voxels); 16 waves, wave w -> voxel w.
// GEMM per voxel: D[b(4 of 16), o(32)] = Scat[b, i(64)] * Wcat[i, o]
// via 16 K-chunks of V_WMMA_F32_16X16X4_F32 x 2 N-tiles, weights via LDS.
__global__ __launch_bounds__(512) void wave_mix_wmma_kernel(
    const float* __restrict__ S,    // [2][8][4][4096][32]
    const float* __restrict__ W1,   // [8][32][32][4096]
    const float* __restrict__ W2,   // [8][32][32][4096]
    float* __restrict__ Mixed)      // [8][4][32][4096]
{
    // staged chunk: 16 i x 32 o x 16 p, row stride 20 dwords (pad 4):
    //   idx = i*640 + o*20 + p
    // 20*o mod 64 injective for o in 0..15 -> conflict-free B reads per half;
    // all strides multiples of 4 dwords -> 16B-aligned b128 LDS traffic.
    __shared__ float lds[16 * 640];   // 40 KiB

    const int tid  = threadIdx.x;
    const int wg   = blockIdx.x;          // 0..2047
    const int k    = wg >> 8;             // subband
    const int p0   = (wg & 255) * 16;     // base voxel
    const int wave = tid >> 5;            // 0..15  -> voxel p0+wave
    const int lane = tid & 31;
    const int m    = lane & 15;           // A/D row (batch; rows >=4 are zero)
    const int h    = lane >> 4;           // half-wave
    const int n    = lane & 15;           // B/D column within N-tile

    // per-wave A base pointers (concat channel blocks: src0 = S1, src1 = S2)
    const float* sA0 = S + ((size_t)((0 * 8 + k) * 4 + m) * 4096 + p0 + wave) * 32;
    const float* sA1 = S + ((size_t)((1 * 8 + k) * 4 + m) * 4096 + p0 + wave) * 32;

    v8f acc0 = {};   // o = 0..15
    v8f acc1 = {};   // o = 16..31

#pragma unroll
    for (int s = 0; s < 4; ++s) {        // 4 stages x 16 concat-channels
        __syncthreads();
        // ---- cooperative, coalesced global->LDS stage (b128 moves) ----
        {
            const float* Wsrc  = (s < 2) ? W1 : W2;
            const int    cbase = (s & 1) * 16;
            const float* Wstage =
                Wsrc + (size_t)((k * 32 + cbase) * 32) * 4096 + p0;
            // 16 i x 32 o x 4 quads = 2048 float4's; 512 threads x 4
#pragma unroll
            for (int it = 0; it < 4; ++it) {
                const int q     = tid + it * 512;
                const int i_loc = q >> 7;          // /(32*4)
                const int rem   = q & 127;
                const int o     = rem >> 2;
                const int pq    = rem & 3;         // quad within 16 voxels
                const float4 v = *(const float4*)(
                    Wstage + (size_t)i_loc * 131072 + o * 4096 + pq * 4);
                *(float4*)&lds[i_loc * 640 + o * 20 + pq * 4] = v;
            }
        }
        __syncthreads();
        // ---- 4 K-chunks of 4 ----
        const float* sA = (s < 2) ? sA0 : sA1;
        const int   cb  = (s & 1) * 16;
#pragma unroll
        for (int kc = 0; kc < 4; ++kc) {
            // A: Scat[b=m, i0..i0+1]; lane layout: VGPR j holds K-local 2h+j
            v2f a; a.x = 0.f; a.y = 0.f;
            if (m < 4) {
                const float* sp = sA + cb + kc * 4 + 2 * h;
                a.x = sp[0];
                a.y = sp[1];
            }
            // B: Wcat[rows 2h+j of chunk, col n], from LDS
            const int il = kc * 4 + 2 * h;
            v2f b0, b1;
            b0.x = lds[(il + 0) * 640 + n * 20 + wave];
            b0.y = lds[(il + 1) * 640 + n * 20 + wave];
            b1.x = lds[(il + 0) * 640 + (16 + n) * 20 + wave];
            b1.y = lds[(il + 1) * 640 + (16 + n) * 20 + wave];

            acc0 = __builtin_amdgcn_wmma_f32_16x16x4_f32(
                false, a, false, b0, (short)0, acc0, false, false);
            acc1 = __builtin_amdgcn_wmma_f32_16x16x4_f32(
                false, a, false, b1, (short)0, acc1, false, false);
        }
    }

    // ---- transpose D through LDS so Mixed stores are coalesced b128 ----
    __syncthreads();
    if (h == 0) {  // D rows b = v + 8h; only b<4 valid -> lanes 0..15, v 0..3
#pragma unroll
        for (int v = 0; v < 4; ++v) {
            lds[(v * 32 + n)      * 20 + wave] = acc0[v];
            lds[(v * 32 + 16 + n) * 20 + wave] = acc1[v];
        }
    }
    __syncthreads();
    {
        // 4 b x 32 o x 4 quads = 512 float4's; one per thread
        const int b  = tid >> 7;
        const int rem = tid & 127;
        const int o  = rem >> 2;
        const int pq = rem & 3;
        const float4 v = *(const float4*)&lds[(b * 32 + o) * 20 + pq * 4];
        *(float4*)&Mixed[(size_t)((k * 4 + b) * 32 + o) * 4096 + p0 + pq * 4] = v;
    }
}

// ------------------------------ Kernel 3: synthesis + upsample --------------
// thread = (b, c, Z, Y, X): reads 8 subband values, writes a 4x4x4 out block.
__global__ __launch_bounds__(256) void wave_inv_kernel(
    const float* __restrict__ Mixed,   // [8][4][32][4096]
    float* __restrict__ out)           // [4][32][64][64][64]
{
    const int t = blockIdx.x * 256 + threadIdx.x;   // 0 .. 2^19-1
    const int X = t & 15, Y = (t >> 4) & 15, Z = (t >> 8) & 15;
    const int c = (t >> 12) & 31, b = t >> 17;
    const int p = t & 4095;

    float M[8];
#pragma unroll
    for (int k = 0; k < 8; ++k)
        M[k] = Mixed[(size_t)((k * 4 + b) * 32 + c) * 4096 + p];

    // inverse sign-Hadamard: V[d] = (1/8) * sum_k (-1)^(d.k) M[k]
    float u1[2][2][2], u2[2][2][2], V[2][2][2];
#pragma unroll
    for (int kz = 0; kz < 2; ++kz)
#pragma unroll
        for (int ky = 0; ky < 2; ++ky) {
            u1[kz][ky][0] = M[kz * 4 + ky * 2 + 0] + M[kz * 4 + ky * 2 + 1];
            u1[kz][ky][1] = M[kz * 4 + ky * 2 + 0] - M[kz * 4 + ky * 2 + 1];
        }
#pragma unroll
    for (int kz = 0; kz < 2; ++kz)
#pragma unroll
        for (int dx = 0; dx < 2; ++dx) {
            u2[kz][0][dx] = u1[kz][0][dx] + u1[kz][1][dx];
            u2[kz][1][dx] = u1[kz][0][dx] - u1[kz][1][dx];
        }
#pragma unroll
    for (int dy = 0; dy < 2; ++dy)
#pragma unroll
        for (int dx = 0; dx < 2; ++dx) {
            V[0][dy][dx] = 0.125f * (u2[0][dy][dx] + u2[1][dy][dx]);
            V[1][dy][dx] = 0.125f * (u2[0][dy][dx] - u2[1][dy][dx]);
        }

    float* ob = out + ((size_t)(b * 32 + c) << 18);
#pragma unroll
    for (int dz = 0; dz < 2; ++dz)
#pragma unroll
        for (int ez = 0; ez < 2; ++ez)
#pragma unroll
            for (int dy = 0; dy < 2; ++dy)
#pragma unroll
                for (int ey = 0; ey < 2; ++ey) {
                    float4 row;
                    row.x = V[dz][dy][0]; row.y = V[dz][dy][0];
                    row.z = V[dz][dy][1]; row.w = V[dz][dy][1];
                    *(float4*)(ob + (((size_t)(4 * Z + 2 * dz + ez) * 64 +
                                      (4 * Y + 2 * dy + ey)) * 64 + 4 * X)) = row;
                }
}

// ---------------------------------------------------------------------------
extern "C" void kernel_launch(void* const* d_in, const int* in_sizes, int n_in,
                              void* d_out, int out_size, void* d_ws, size_t ws_size,
                              hipStream_t stream) {
    const float* x1 = (const float*)d_in[0];
    const float* x2 = (const float*)d_in[1];
    const float* W1 = (const float*)d_in[2];
    const float* W2 = (const float*)d_in[3];
    float* out = (float*)d_out;

    float* S     = (float*)d_ws;                       // 2*8*4*4096*32 = 8M floats
    float* Mixed = S + (size_t)2 * 8 * 4 * 4096 * 32;  // 8*4*32*4096  = 4M floats

    wave_dwt_kernel<<<4096, 256, 0, stream>>>(x1, x2, S);
    wave_mix_wmma_kernel<<<2048, 512, 0, stream>>>(S, W1, W2, Mixed);
    wave_inv_kernel<<<2048, 256, 0, stream>>>(Mixed, out);
}